// ConvNearestNeightbor_52905407152506
// MI455X (gfx1250) — compile-verified
//
#include <hip/hip_runtime.h>
#include <stdint.h>
#include <math.h>

// Problem constants from the reference: x (B=16, C=32, H=32, W=32),
// neighbors (NUM=32, C=32, 9), out (B, NUM*C, H, W) fp32.
#define B_  16
#define C_  32
#define H_  32
#define W_  32
#define NUM_ 32
#define PH_ 34   // padded rows (1 top + 32 + 1 bottom)
#define PW_ 35   // padded row pitch (1 + 32 + 1, +1 to de-skew LDS banks)

#if defined(__AMDGCN__) && __has_builtin(__builtin_amdgcn_global_load_async_to_lds_b32)
#define USE_ASYNC_LDS 1
typedef __attribute__((address_space(1))) int g_int_t;   // global int*
typedef __attribute__((address_space(3))) int l_int_t;   // LDS int*
#else
#define USE_ASYNC_LDS 0
#endif

__global__ __launch_bounds__(256) void convnn_kernel(
    const float* __restrict__ x,    // (B, C, H, W)
    const float* __restrict__ nb,   // (NUM, C, 9)
    float* __restrict__ out)        // (B, NUM*C, H, W)
{
  __shared__ float xp[PH_ * PW_];    // zero-padded plane for this (b,c)
  __shared__ float nbl[NUM_ * 9];    // neighbors[:, c, :]

  const int bc  = blockIdx.x;        // b*C + c
  const int b   = bc >> 5;           // bc / C_
  const int c   = bc & 31;           // bc % C_
  const int tid = threadIdx.x;       // 0..255

  // ---- stage the 32x32 interior of x into LDS (async, ASYNCcnt-tracked) ----
  const float* xplane = x + (size_t)bc * (H_ * W_);
  {
    const int h = tid >> 5;          // 0..7
    const int w = tid & 31;
#if USE_ASYNC_LDS
#pragma unroll
    for (int j = 0; j < 4; ++j) {    // rows h, h+8, h+16, h+24
      const int hh = h + 8 * j;
      __builtin_amdgcn_global_load_async_to_lds_b32(
          (g_int_t*)(xplane + hh * W_ + w),
          (l_int_t*)&xp[(hh + 1) * PW_ + (w + 1)],
          /*offset=*/0, /*cpol=*/0);
    }
#else
#pragma unroll
    for (int j = 0; j < 4; ++j) {
      const int hh = h + 8 * j;
      xp[(hh + 1) * PW_ + (w + 1)] = xplane[hh * W_ + w];
    }
#endif
  }

  // ---- zero the 132 padding-border cells (disjoint from async interior) ----
  // rows 0 and 33 (34 cells each), cols 0 and 33 of rows 1..32 (32 cells each)
  if (tid < 132) {
    int idx;
    if (tid < 34)        idx = tid;                          // top row
    else if (tid < 68)   idx = 33 * PW_ + (tid - 34);        // bottom row
    else if (tid < 100)  idx = (tid - 67) * PW_;             // left col, rows 1..32
    else                 idx = (tid - 99) * PW_ + 33;        // right col, rows 1..32
    xp[idx] = 0.0f;
  }

  // ---- stage neighbors[:, c, :] (288 floats) into LDS ----
  {
    const int i = tid;                                       // 0..255
    const int n = i / 9, k = i - n * 9;
    nbl[i] = nb[n * (C_ * 9) + c * 9 + k];
    if (tid < NUM_ * 9 - 256) {                              // remaining 32
      const int i2 = tid + 256;
      const int n2 = i2 / 9, k2 = i2 - n2 * 9;
      nbl[i2] = nb[n2 * (C_ * 9) + c * 9 + k2];
    }
  }

#if USE_ASYNC_LDS
  // Drain this wave's async LDS writes, then barrier so all waves' data is visible.
#if __has_builtin(__builtin_amdgcn_s_wait_asynccnt)
  __builtin_amdgcn_s_wait_asynccnt(0);
#else
  asm volatile("s_wait_asynccnt 0" ::: "memory");
#endif
#endif
  __syncthreads();

  // ---- hoist each thread's 4 pixel windows into registers ----
  // pixel j: (h0 + 8j, w0); reference shift k=(kr*3+kc) reads x[h-row, w-col]
  //   = xp[h + 2 - kr][w + 2 - kc]
  const int h0 = tid >> 5;
  const int w0 = tid & 31;
  float xv[4][9];
#pragma unroll
  for (int j = 0; j < 4; ++j) {
    const int h = h0 + 8 * j;
#pragma unroll
    for (int kr = 0; kr < 3; ++kr)
#pragma unroll
      for (int kc = 0; kc < 3; ++kc)
        xv[j][kr * 3 + kc] = xp[(h + 2 - kr) * PW_ + (w0 + 2 - kc)];
  }

  // ---- main loop: 32 codebook entries, 4 pixels each ----
  const size_t outbase = ((size_t)b * (NUM_ * C_) + c) * (size_t)(H_ * W_);
  for (int n = 0; n < NUM_; ++n) {
    float nv[9];
#pragma unroll
    for (int k = 0; k < 9; ++k) nv[k] = nbl[n * 9 + k];  // LDS broadcast reads

    float* op = out + outbase + (size_t)n * (C_ * H_ * W_);
#pragma unroll
    for (int j = 0; j < 4; ++j) {
      float m = fabsf(xv[j][0] - nv[0]);
#pragma unroll
      for (int k = 1; k < 9; ++k) m = fmaxf(m, fabsf(xv[j][k] - nv[k]));
      const int h = h0 + 8 * j;
      // 64 MB write-once stream: non-temporal store, coalesced per wave row.
      __builtin_nontemporal_store(m, op + h * W_ + w0);
    }
  }
}

extern "C" void kernel_launch(void* const* d_in, const int* in_sizes, int n_in,
                              void* d_out, int out_size, void* d_ws, size_t ws_size,
                              hipStream_t stream) {
  (void)in_sizes; (void)n_in; (void)d_ws; (void)ws_size; (void)out_size;
  const float* x  = (const float*)d_in[0];
  const float* nb = (const float*)d_in[1];
  float* out      = (float*)d_out;
  convnn_kernel<<<dim3(B_ * C_), dim3(256), 0, stream>>>(x, nb, out);
}